// E3SchNetInteraction_53171695125405
// MI455X (gfx1250) — compile-verified
//
#include <hip/hip_runtime.h>
#include <math.h>

typedef float v2f __attribute__((ext_vector_type(2)));
typedef float v8f __attribute__((ext_vector_type(8)));

#define N_NODES 16384
#define E_EDGES 262144

// D = A(16x4,f32) * B(4x16,f32) + C(16x16,f32)
__device__ __forceinline__ v8f wmma4(v2f a, v2f b, v8f c) {
  return __builtin_amdgcn_wmma_f32_16x16x4_f32(
      /*neg_a=*/false, a, /*neg_b=*/false, b,
      /*c_mod=*/(short)0, c, /*reuse_a=*/false, /*reuse_b=*/false);
}

// shifted softplus: log(1+exp(x)) - log(2), numerically stable
__device__ __forceinline__ float ssp_f(float x) {
  float ax = fabsf(x);
  return fmaxf(x, 0.0f) + log1pf(__expf(-ax)) - 0.69314718055994531f;
}

// ---------------------------------------------------------------------------
// Kernel 1: node embedding into interleaved layout
//   h_all[n][f][0] = (xs @ w_s)[n][f] / sqrt(128)
//   h_all[n][f][1+c] = (xv[:,:,c] @ w_v)[n][f] / sqrt(128)
// One wave per (16-node tile, 16-col tile, channel {s,vx,vy,vz}).
// Branch-free inner loop via uniform base pointer + stride.
// ---------------------------------------------------------------------------
__global__ void node_embed_kernel(const float* __restrict__ x,
                                  const float* __restrict__ w_s,
                                  const float* __restrict__ w_v,
                                  float* __restrict__ h_all) {
  const int lane = threadIdx.x & 31;
  const int wave = threadIdx.x >> 5;
  const int gid  = blockIdx.x * 4 + wave;           // 8192 wave-tiles total
  const int node_tile = gid >> 3;
  const int rest = gid & 7;
  const int ch = rest >> 1;                         // 0 = scalar, 1..3 = vec comp
  const int ct = rest & 1;                          // column tile (0 or 1)
  const int nb = node_tile * 16;
  const int m  = lane & 15;
  const int hi = lane >> 4;
  const int koff = hi * 2;

  // A: row (nb+m); scalar path reads x[:, k], vector reads x[:, 128 + 3k + c]
  const float* arow = x + (size_t)(nb + m) * 512 + (ch ? (128 + ch - 1) : 0);
  const int astr = ch ? 3 : 1;
  // B: column (ct*16 + m); row stride 32
  const float* bcol = (ch ? w_v : w_s) + ct * 16 + m;

  v8f acc = {};
#pragma unroll
  for (int ks = 0; ks < 128; ks += 4) {
    const int k0 = ks + koff;
    v2f a, b;
    a.x = arow[k0 * astr];
    a.y = arow[(k0 + 1) * astr];
    b.x = bcol[k0 * 32];
    b.y = bcol[(k0 + 1) * 32];
    acc = wmma4(a, b, acc);
  }

  const float scale = 0.088388347648318447f;        // 1/sqrt(128)
  const int col = ct * 16 + m;
  // dest: h_all[(nb+mm)*128 + col*4 + comp], comp = 0 (scalar) or 1+c
  float* dst = h_all + (size_t)nb * 128 + col * 4 + (ch ? ch : 0) + (ch ? 0 : 0);
  dst = h_all + (size_t)nb * 128 + col * 4 + ch;    // ch==0 -> comp 0; ch=1..3 -> comp 1..3
#pragma unroll
  for (int r = 0; r < 8; ++r) {
    const int mm = r + 8 * hi;
    dst[(size_t)mm * 128] = acc[r] * scale;
  }
}

// ---------------------------------------------------------------------------
// Kernel 2: edge filter MLP (WMMA) + edge tensor products + scatter-add.
// 4 waves/block, 16 edges per wave.  Only the 4 live filter channels (of 6)
// are kept; t_pv / t_q aggregates are dead in the reference and skipped.
// ---------------------------------------------------------------------------
__global__ void edge_kernel(const float* __restrict__ f_ij,
                            const float* __restrict__ rcut,
                            const float* __restrict__ Yr,
                            const int*   __restrict__ idx_i,
                            const int*   __restrict__ idx_j,
                            const float* __restrict__ W1,
                            const float* __restrict__ b1,
                            const float* __restrict__ W2,
                            const float* __restrict__ b2,
                            const float* __restrict__ h_all,
                            float* __restrict__ agg) {
  __shared__ float sh_h[4][16][32];       // 8 KB
  __shared__ float sh_Wc[4][16][32][4];   // 32 KB (live channels 0..3 only)

  const int lane = threadIdx.x & 31;
  const int wave = threadIdx.x >> 5;
  const int eb   = blockIdx.x * 64 + wave * 16;
  const int m  = lane & 15;
  const int hi = lane >> 4;
  const int koff = hi * 2;

  // --- GEMM1: h = ssp(f_ij @ W1 + b1)   (16x20 @ 20x32, K = 5 steps of 4) ---
  const float* frow = f_ij + (size_t)(eb + m) * 20;
#pragma unroll
  for (int t = 0; t < 2; ++t) {
    const float* bcol = W1 + t * 16 + m;
    v8f acc = {};
#pragma unroll
    for (int ks = 0; ks < 20; ks += 4) {
      const int k0 = ks + koff;
      v2f a, b;
      a.x = frow[k0];
      a.y = frow[k0 + 1];
      b.x = bcol[k0 * 32];
      b.y = bcol[(k0 + 1) * 32];
      acc = wmma4(a, b, acc);
    }
    const float bias = b1[t * 16 + m];
#pragma unroll
    for (int r = 0; r < 8; ++r) {
      const int mm = r + 8 * hi;
      sh_h[wave][mm][t * 16 + m] = ssp_f(acc[r] + bias);
    }
  }
  __syncthreads();

  // --- GEMM2: Wij = (h @ W2 + b2) * rcut   (16x32 @ 32x192, 12 col tiles) ---
  for (int t = 0; t < 12; ++t) {
    const int col = t * 16 + m;          // 0..191
    const float* bcol = W2 + col;
    v8f acc = {};
#pragma unroll
    for (int ks = 0; ks < 32; ks += 4) {
      const int k0 = ks + koff;
      v2f a, b;
      a.x = sh_h[wave][m][k0];
      a.y = sh_h[wave][m][k0 + 1];
      b.x = bcol[k0 * 192];
      b.y = bcol[(k0 + 1) * 192];
      acc = wmma4(a, b, acc);
    }
    const int f   = col / 6;
    const int chn = col % 6;
    const float bias = b2[col];
#pragma unroll
    for (int r = 0; r < 8; ++r) {
      const int mm = r + 8 * hi;
      const float val = (acc[r] + bias) * rcut[eb + mm];
      if (chn < 4) sh_Wc[wave][mm][f][chn] = val;
    }
  }
  __syncthreads();

  // --- edge products + scatter: one edge per iteration, 32 features/lanes ---
  const float IS3 = 0.57735026918962576f;
  const int f = lane;                     // feature 0..31
#pragma unroll 2
  for (int it = 0; it < 16; ++it) {
    const int edge = eb + it;
    const int j = idx_j[edge];
    const int i = idx_i[edge];
    const float ys = Yr[(size_t)edge * 4 + 0];
    const float yx = Yr[(size_t)edge * 4 + 1];
    const float yy = Yr[(size_t)edge * 4 + 2];
    const float yz = Yr[(size_t)edge * 4 + 3];
    // single 128-bit gather: {s, vx, vy, vz}
    const float4 h = *(const float4*)(h_all + ((size_t)j * 32 + f) * 4);
    const float s = h.x, vx = h.y, vy = h.z, vz = h.w;
    const float W0 = sh_Wc[wave][it][f][0];
    const float Wa = sh_Wc[wave][it][f][1];
    const float Wb = sh_Wc[wave][it][f][2];
    const float Wc = sh_Wc[wave][it][f][3];

    const float c0 = s * ys * W0;                                  // t_s1
    const float c1 = (vx * yx + vy * yy + vz * yz) * IS3 * Wa;     // t_s2
    const float sb = s * Wb;                                       // t_v1 factor
    const float vc = ys * Wc;                                      // t_v2 factor
    float* dst = agg + (size_t)i * 256 + f * 8;
    __hip_atomic_fetch_add(dst + 0, c0,      __ATOMIC_RELAXED, __HIP_MEMORY_SCOPE_AGENT);
    __hip_atomic_fetch_add(dst + 1, c1,      __ATOMIC_RELAXED, __HIP_MEMORY_SCOPE_AGENT);
    __hip_atomic_fetch_add(dst + 2, sb * yx, __ATOMIC_RELAXED, __HIP_MEMORY_SCOPE_AGENT);
    __hip_atomic_fetch_add(dst + 3, sb * yy, __ATOMIC_RELAXED, __HIP_MEMORY_SCOPE_AGENT);
    __hip_atomic_fetch_add(dst + 4, sb * yz, __ATOMIC_RELAXED, __HIP_MEMORY_SCOPE_AGENT);
    __hip_atomic_fetch_add(dst + 5, vx * vc, __ATOMIC_RELAXED, __HIP_MEMORY_SCOPE_AGENT);
    __hip_atomic_fetch_add(dst + 6, vy * vc, __ATOMIC_RELAXED, __HIP_MEMORY_SCOPE_AGENT);
    __hip_atomic_fetch_add(dst + 7, vz * vc, __ATOMIC_RELAXED, __HIP_MEMORY_SCOPE_AGENT);
  }
}

// ---------------------------------------------------------------------------
// Kernel 3: output MLPs.  wave0 = scalar path (with ssp), waves 1..3 = vector
// components.  GEMM1 K=64 from agg (branch-free split loops), GEMM2 K=128 via
// LDS, writes d_out directly.
// ---------------------------------------------------------------------------
__global__ void output_kernel(const float* __restrict__ agg,
                              const float* __restrict__ w1s,
                              const float* __restrict__ w1v,
                              const float* __restrict__ w2s,
                              const float* __restrict__ w2v,
                              float* __restrict__ out) {
  __shared__ float sh_mid[4][16][128];    // 32 KB
  const int lane = threadIdx.x & 31;
  const int wave = threadIdx.x >> 5;
  const int nb = blockIdx.x * 16;
  const int m  = lane & 15;
  const int hi = lane >> 4;
  const int koff = hi * 2;
  const int c = wave - 1;
  const float* Wa = (wave == 0) ? w1s : w1v;
  const float* Wb = (wave == 0) ? w2s : w2v;
  const int ch_lo = (wave == 0) ? 0 : 2 + c;   // agg channel for k in [0,32)
  const int ch_hi = (wave == 0) ? 1 : 5 + c;   // agg channel for k in [32,64)

  // A row base: agg[(nb+m)*256 + k*8 + ch]
  const float* arow_lo = agg + (size_t)(nb + m) * 256 + ch_lo;
  const float* arow_hi = agg + (size_t)(nb + m) * 256 + ch_hi;

  // --- GEMM1: (16x64) @ (64x128), scale 1/8, ssp on scalar path ---
  for (int t = 0; t < 8; ++t) {
    const float* bcol = Wa + t * 16 + m;
    v8f acc = {};
#pragma unroll
    for (int ks = 0; ks < 32; ks += 4) {         // k in [0,32): channel ch_lo
      const int k0 = ks + koff;
      v2f a, b;
      a.x = arow_lo[k0 * 8];
      a.y = arow_lo[(k0 + 1) * 8];
      b.x = bcol[k0 * 128];
      b.y = bcol[(k0 + 1) * 128];
      acc = wmma4(a, b, acc);
    }
#pragma unroll
    for (int ks = 0; ks < 32; ks += 4) {         // k in [32,64): channel ch_hi
      const int k0 = ks + koff;
      v2f a, b;
      a.x = arow_hi[k0 * 8];
      a.y = arow_hi[(k0 + 1) * 8];
      b.x = bcol[(k0 + 32) * 128];
      b.y = bcol[(k0 + 33) * 128];
      acc = wmma4(a, b, acc);
    }
#pragma unroll
    for (int r = 0; r < 8; ++r) {
      const int mm = r + 8 * hi;
      float val = acc[r] * 0.125f;                 // 1/sqrt(64)
      if (wave == 0) val = ssp_f(val);
      sh_mid[wave][mm][t * 16 + m] = val;
    }
  }
  __syncthreads();

  // --- GEMM2: (16x128) @ (128x128), scale 1/sqrt(128), write out ---
  for (int t = 0; t < 8; ++t) {
    const float* bcol = Wb + t * 16 + m;
    v8f acc = {};
#pragma unroll
    for (int ks = 0; ks < 128; ks += 4) {
      const int k0 = ks + koff;
      v2f a, b;
      a.x = sh_mid[wave][m][k0];
      a.y = sh_mid[wave][m][k0 + 1];
      b.x = bcol[k0 * 128];
      b.y = bcol[(k0 + 1) * 128];
      acc = wmma4(a, b, acc);
    }
    const int col = t * 16 + m;
    // scalar path -> out[:, col]; vector path -> out[:, 128 + col*3 + c]
    float* dst = (wave == 0) ? (out + (size_t)nb * 512 + col)
                             : (out + (size_t)nb * 512 + 128 + col * 3 + c);
#pragma unroll
    for (int r = 0; r < 8; ++r) {
      const int mm = r + 8 * hi;
      dst[(size_t)mm * 512] = acc[r] * 0.088388347648318447f;
    }
  }
}

// ---------------------------------------------------------------------------
extern "C" void kernel_launch(void* const* d_in, const int* in_sizes, int n_in,
                              void* d_out, int out_size, void* d_ws, size_t ws_size,
                              hipStream_t stream) {
  const float* x        = (const float*)d_in[0];
  const int*   idx_i    = (const int*)d_in[1];
  const int*   idx_j    = (const int*)d_in[2];
  const float* f_ij     = (const float*)d_in[3];
  const float* rcut     = (const float*)d_in[4];
  const float* Yr       = (const float*)d_in[5];
  const float* w_in2f_s = (const float*)d_in[6];
  const float* w_in2f_v = (const float*)d_in[7];
  const float* W1       = (const float*)d_in[8];
  const float* b1       = (const float*)d_in[9];
  const float* W2       = (const float*)d_in[10];
  const float* b2       = (const float*)d_in[11];
  const float* w_f2o1_s = (const float*)d_in[12];
  const float* w_f2o1_v = (const float*)d_in[13];
  const float* w_f2o2_s = (const float*)d_in[14];
  const float* w_f2o2_v = (const float*)d_in[15];

  float* h_all = (float*)d_ws;                         // N*128 floats (8 MB)
  float* agg   = h_all + (size_t)N_NODES * 128;        // N*256 floats (16 MB)

  hipMemsetAsync(agg, 0, (size_t)N_NODES * 256 * sizeof(float), stream);

  node_embed_kernel<<<2048, 128, 0, stream>>>(x, w_in2f_s, w_in2f_v, h_all);
  edge_kernel<<<E_EDGES / 64, 128, 0, stream>>>(f_ij, rcut, Yr, idx_i, idx_j,
                                                W1, b1, W2, b2, h_all, agg);
  output_kernel<<<N_NODES / 16, 128, 0, stream>>>(agg, w_f2o1_s, w_f2o1_v,
                                                  w_f2o2_s, w_f2o2_v,
                                                  (float*)d_out);
}